// BaseModel_64424509440417
// MI455X (gfx1250) — compile-verified
//
#include <hip/hip_runtime.h>
#include <float.h>

typedef __attribute__((ext_vector_type(2))) float v2f;
typedef __attribute__((ext_vector_type(8))) float v8f;

#define NPTS   16384
#define TILE   16
#define NTILE  (NPTS / TILE)   // 1024 tiles per axis
#define EPS    1e-7f

// ---------------------------------------------------------------------------
// Kernel A: apply per-frame pose to each point; emit padded (x,y,z,0) points
// and squared norms for both est and gt transforms.
// ---------------------------------------------------------------------------
__global__ __launch_bounds__(256)
void transform_kernel(const float* __restrict__ pts,
                      const int*   __restrict__ tidx,
                      const float* __restrict__ est_poses,
                      const float* __restrict__ gt_poses,
                      float* __restrict__ gt4,  float* __restrict__ est4,
                      float* __restrict__ gtsq, float* __restrict__ estsq)
{
    int n = blockIdx.x * blockDim.x + threadIdx.x;
    if (n >= NPTS) return;

    float px = pts[3 * n + 0];
    float py = pts[3 * n + 1];
    float pz = pts[3 * n + 2];
    int   t  = tidx[n];

    const float* E = est_poses + t * 16;   // row-major 4x4
    const float* G = gt_poses  + t * 16;

    float ex = fmaf(E[0], px, fmaf(E[1], py, fmaf(E[2],  pz, E[3])));
    float ey = fmaf(E[4], px, fmaf(E[5], py, fmaf(E[6],  pz, E[7])));
    float ez = fmaf(E[8], px, fmaf(E[9], py, fmaf(E[10], pz, E[11])));
    float gx = fmaf(G[0], px, fmaf(G[1], py, fmaf(G[2],  pz, G[3])));
    float gy = fmaf(G[4], px, fmaf(G[5], py, fmaf(G[6],  pz, G[7])));
    float gz = fmaf(G[8], px, fmaf(G[9], py, fmaf(G[10], pz, G[11])));

    est4[4 * n + 0] = ex;  est4[4 * n + 1] = ey;
    est4[4 * n + 2] = ez;  est4[4 * n + 3] = 0.0f;
    gt4 [4 * n + 0] = gx;  gt4 [4 * n + 1] = gy;
    gt4 [4 * n + 2] = gz;  gt4 [4 * n + 3] = 0.0f;

    estsq[n] = fmaf(ex, ex, fmaf(ey, ey, ez * ez));
    gtsq [n] = fmaf(gx, gx, fmaf(gy, gy, gz * gz));
}

// ---------------------------------------------------------------------------
// Kernel B: for each row i of point-set A, compute
//     outmin[i] = min_j ( Asq[i] + Bsq[j] - 2 * dot(A_i, B_j) )
// One workgroup (8 waves) per 16-row block; each wave streams 1/8 of the
// 1024 column tiles with one v_wmma_f32_16x16x4_f32 per 16x16 tile.
//
// Trick: A fragment is pre-scaled by -2 once, and the accumulator is seeded
// with C[v] = Asq[row] + Bsq[col], so the WMMA emits the finished d2 tile:
//     D = (-2A) * B + (Asq + Bsq)
//
// f32 WMMA fragment layouts (wave32):
//   A 16x4 : lane m (0-15) VGPR{0,1} = A[m][K=0,1]; lanes 16-31 = K=2,3
//   B 4x16 : lane m (0-15) VGPR{0,1} = B[K=0,1][m]; lanes 16-31 = K=2,3
//   C/D    : VGPR v, lanes 0-15 -> (M=v,   N=lane),
//                    lanes 16-31 -> (M=v+8, N=lane-16)
// Points stored as float4 (x,y,z,0): fragment load is one b64 per lane.
// ---------------------------------------------------------------------------
__global__ __launch_bounds__(256)
void chamfer_min_kernel(const float* __restrict__ A4, const float* __restrict__ Asq,
                        const float* __restrict__ B4, const float* __restrict__ Bsq,
                        float* __restrict__ outmin)
{
    __shared__ float lds[8][16];

    const int i0   = blockIdx.x * TILE;
    const int lane = threadIdx.x & 31;
    const int wave = threadIdx.x >> 5;
    const int half = lane >> 4;     // 0: K=0,1 / rows 0-7   1: K=2,3 / rows 8-15
    const int m    = lane & 15;

    const v2f* A2 = (const v2f*)A4;
    const v2f* B2 = (const v2f*)B4;

    // A fragment is invariant over the whole column sweep: load once and
    // fold the Chamfer -2 factor into it (D = -2*A*B + C).
    v2f afrag = A2[(i0 + m) * 2 + half] * (-2.0f);

    // Row squared-norms needed by this lane's D elements (fixed for the block).
    float asq[8];
#pragma unroll
    for (int v = 0; v < 8; ++v) asq[v] = Asq[i0 + v + 8 * half];

    // Per-lane running column-partial row mins (reduced across lanes later).
    float rowpart[8];
#pragma unroll
    for (int v = 0; v < 8; ++v) rowpart[v] = FLT_MAX;

    // Constant trip count: scalar loop control, EXEC stays all-1s (required
    // for WMMA), and the unroller can pipeline loads against the WMMAs.
#pragma unroll 4
    for (int k = 0; k < NTILE / 8; ++k) {
        const int j0 = (wave + 8 * k) * TILE;
        v2f   bfrag = B2[(j0 + m) * 2 + half];
        float bsq   = Bsq[j0 + m];

        v8f c;
#pragma unroll
        for (int v = 0; v < 8; ++v) c[v] = asq[v] + bsq;

        v8f d = __builtin_amdgcn_wmma_f32_16x16x4_f32(
            /*neg_a=*/false, afrag, /*neg_b=*/false, bfrag,
            /*c_mod=*/(short)0, c, /*reuse_a=*/false, /*reuse_b=*/false);

#pragma unroll
        for (int v = 0; v < 8; ++v)
            rowpart[v] = fminf(rowpart[v], d[v]);
    }

    // Reduce each row across the 16 lanes of its half-wave (xor tree stays
    // within the half since all masks < 16).
#pragma unroll
    for (int v = 0; v < 8; ++v) {
        float r = rowpart[v];
        r = fminf(r, __shfl_xor(r, 1, 32));
        r = fminf(r, __shfl_xor(r, 2, 32));
        r = fminf(r, __shfl_xor(r, 4, 32));
        r = fminf(r, __shfl_xor(r, 8, 32));
        rowpart[v] = r;
    }

    if (m == 0) {                       // lane 0 (rows 0-7) and lane 16 (rows 8-15)
#pragma unroll
        for (int v = 0; v < 8; ++v) lds[wave][half * 8 + v] = rowpart[v];
    }
    __syncthreads();

    if (threadIdx.x < 16) {             // combine 8 waves; exclusive row write
        float r = lds[0][threadIdx.x];
#pragma unroll
        for (int w = 1; w < 8; ++w) r = fminf(r, lds[w][threadIdx.x]);
        outmin[i0 + threadIdx.x] = r;
    }
}

// ---------------------------------------------------------------------------
// Kernel C: deterministic single-block weighted reduction.
//   chamfer = 0.5 * (sum ind*dist1 + sum ind*dist2) / (count + eps)
//   l2      = sum ind*|est-gt| / (count + eps)
// ---------------------------------------------------------------------------
__global__ __launch_bounds__(256)
void finalize_kernel(const int*   __restrict__ tidx,
                     const float* __restrict__ gt4, const float* __restrict__ est4,
                     const float* __restrict__ dist1, const float* __restrict__ dist2,
                     float* __restrict__ out)
{
    __shared__ float s1s[256], s2s[256], cs[256], l2s[256];

    float s1 = 0.0f, s2 = 0.0f, cnt = 0.0f, l2 = 0.0f;
    for (int i = threadIdx.x; i < NPTS; i += 256) {
        float ind = (tidx[i] == 1) ? 1.0f : 0.0f;
        s1  += ind * dist1[i];
        s2  += ind * dist2[i];
        cnt += ind;
        float dx = est4[4 * i + 0] - gt4[4 * i + 0];
        float dy = est4[4 * i + 1] - gt4[4 * i + 1];
        float dz = est4[4 * i + 2] - gt4[4 * i + 2];
        l2 += ind * sqrtf(fmaf(dx, dx, fmaf(dy, dy, dz * dz)));
    }
    int tid = threadIdx.x;
    s1s[tid] = s1; s2s[tid] = s2; cs[tid] = cnt; l2s[tid] = l2;
    __syncthreads();
    for (int s = 128; s > 0; s >>= 1) {
        if (tid < s) {
            s1s[tid] += s1s[tid + s];
            s2s[tid] += s2s[tid + s];
            cs [tid] += cs [tid + s];
            l2s[tid] += l2s[tid + s];
        }
        __syncthreads();
    }
    if (tid == 0) {
        float wden = cs[0] + EPS;
        out[0] = 0.5f * (s1s[0] + s2s[0]) / wden;   // chamfer
        out[1] = l2s[0] / wden;                     // l2
    }
}

// ---------------------------------------------------------------------------
// Workspace layout (floats): GT4[4N] | EST4[4N] | GTSQ[N] | ESTSQ[N] | D1[N] | D2[N]
// ---------------------------------------------------------------------------
extern "C" void kernel_launch(void* const* d_in, const int* in_sizes, int n_in,
                              void* d_out, int out_size, void* d_ws, size_t ws_size,
                              hipStream_t stream) {
    const float* pts       = (const float*)d_in[0];
    const int*   tidx      = (const int*)  d_in[1];
    const float* est_poses = (const float*)d_in[2];
    const float* gt_poses  = (const float*)d_in[3];

    float* ws    = (float*)d_ws;
    float* GT4   = ws;
    float* EST4  = ws + 4 * NPTS;
    float* GTSQ  = ws + 8 * NPTS;
    float* ESTSQ = ws + 9 * NPTS;
    float* D1    = ws + 10 * NPTS;
    float* D2    = ws + 11 * NPTS;

    transform_kernel<<<NPTS / 256, 256, 0, stream>>>(
        pts, tidx, est_poses, gt_poses, GT4, EST4, GTSQ, ESTSQ);

    // dist1[i] = min_j d2(gt_i, est_j)
    chamfer_min_kernel<<<NTILE, 256, 0, stream>>>(GT4, GTSQ, EST4, ESTSQ, D1);
    // dist2[j] = min_i d2(gt_i, est_j)  (same computation, roles swapped)
    chamfer_min_kernel<<<NTILE, 256, 0, stream>>>(EST4, ESTSQ, GT4, GTSQ, D2);

    finalize_kernel<<<1, 256, 0, stream>>>(tidx, GT4, EST4, D1, D2, (float*)d_out);
}